// Memcodes_40235253629346
// MI455X (gfx1250) — compile-verified
//
#include <hip/hip_runtime.h>

typedef float v2f __attribute__((ext_vector_type(2)));
typedef float v8f __attribute__((ext_vector_type(8)));

#define B_ 4
#define N_ 1024
#define D_ 1024
#define H_ 16
#define C_ 1024
#define d_ 64
#define WAVES_PER_BLOCK 8

// ---------------------------------------------------------------------------
// Phase 1: k = codes @ wk  (packed layout (H, d/2, C, 2) for WMMA B-fragments)
//          v = codes @ wv  (row-major (H, C, d) for the gather)
// One thread per output element; trivially small vs. the 256MB gumbel stream.
// ---------------------------------------------------------------------------
__global__ void __launch_bounds__(256)
memcodes_kv_kernel(const float* __restrict__ codes,
                   const float* __restrict__ wk,
                   const float* __restrict__ wv,
                   float* __restrict__ kp,   // (H, d/2, C, 2)
                   float* __restrict__ v)    // (H, C, d)
{
    int t = blockIdx.x * blockDim.x + threadIdx.x;   // H*C*d = 1,048,576
    int h   = t >> 16;          // / (C_*d_)
    int rem = t & 65535;
    int c   = rem >> 6;
    int kk  = rem & 63;

    const float* crow = codes + ((size_t)h * C_ + c) * d_;
    const float* wkh  = wk + (size_t)h * d_ * d_;
    const float* wvh  = wv + (size_t)h * d_ * d_;

    float sk = 0.f, sv = 0.f;
#pragma unroll 8
    for (int i = 0; i < d_; ++i) {
        float ci = crow[i];
        sk = fmaf(ci, wkh[i * d_ + kk], sk);
        sv = fmaf(ci, wvh[i * d_ + kk], sv);
    }
    v[((size_t)h * C_ + c) * d_ + kk] = sv;
    // packed: element (kk, c) at [h][kk/2][c][kk&1]
    kp[(size_t)h * (d_ / 2) * C_ * 2 + (size_t)(kk >> 1) * (C_ * 2) + c * 2 + (kk & 1)] = sk;
}

// ---------------------------------------------------------------------------
// Phase 2: per wave32, one 16-row N-tile of one (b,h):
//   logits(16x1024) via V_WMMA_F32_16X16X4_F32, + gumbel, running argmax,
//   shuffle-reduce, write idx, gather v rows to out.
// ---------------------------------------------------------------------------
__global__ void __launch_bounds__(WAVES_PER_BLOCK * 32)
memcodes_main_kernel(const float* __restrict__ x,
                     const float* __restrict__ gumbel,
                     const float* __restrict__ kp,    // (H, d/2, C, 2)
                     const float* __restrict__ v,     // (H, C, d)
                     float* __restrict__ out,         // (B, N, D)
                     int* __restrict__ idx_out)       // (B, H, N)
{
    const int lane = threadIdx.x & 31;
    const int wave = threadIdx.x >> 5;
    const int tile = blockIdx.x * WAVES_PER_BLOCK + wave;   // 0..4095

    const int bh = tile >> 6;       // 64 tiles per (b,h)
    const int nt = tile & 63;
    const int b  = bh >> 4;
    const int h  = bh & 15;
    const int n0 = nt << 4;

    const int half = lane >> 4;     // 0: lanes 0-15, 1: lanes 16-31
    const int l15  = lane & 15;

    // ---- A fragments: q tile 16x64, scaled by d^-0.5 = 0.125 ----
    // ISA 16x4 f32 A layout: lane m<16 holds (K0,K1); lane 16+m holds (K2,K3)
    const float* qrow = x + ((size_t)(b * N_ + n0 + l15)) * D_ + h * d_;
    v2f a[16];
#pragma unroll
    for (int j = 0; j < 16; ++j) {
        int kk = 4 * j + 2 * half;
        a[j].x = qrow[kk]     * 0.125f;
        a[j].y = qrow[kk + 1] * 0.125f;
    }

    const float* kph   = kp + (size_t)h * (d_ / 2) * C_ * 2;
    const float* gbase = gumbel + ((size_t)((b * H_ + h) * N_ + n0)) * C_;

    float best[8];
    int   bidx[8];
#pragma unroll
    for (int r = 0; r < 8; ++r) { best[r] = -3.402823466e38f; bidx[r] = 0; }

    for (int c0 = 0; c0 < C_; c0 += 16) {
        v8f acc = {0.f, 0.f, 0.f, 0.f, 0.f, 0.f, 0.f, 0.f};
        // B layout mirrors A: lane n<16 holds (K0,K1) of column n; 16+n holds (K2,K3)
        const float* bp = kph + (size_t)half * (2 * C_) + (size_t)(c0 + l15) * 2;
#pragma unroll
        for (int j = 0; j < 16; ++j) {
            v2f bf = *(const v2f*)(bp + (size_t)(2 * j) * (2 * C_));
            acc = __builtin_amdgcn_wmma_f32_16x16x4_f32(
                false, a[j], false, bf, (short)0, acc, false, false);
        }
        // D layout: VGPR r, lane L -> M = r + 8*half, N = c0 + l15
#pragma unroll
        for (int r = 0; r < 8; ++r) {
            int   m   = r + half * 8;
            float g   = gbase[(size_t)m * C_ + c0 + l15];
            float val = acc[r] + g;
            int   ci  = c0 + l15;
            bool take = (val > best[r]) | ((val == best[r]) & (ci < bidx[r]));
            best[r] = take ? val : best[r];
            bidx[r] = take ? ci  : bidx[r];
        }
    }

    // ---- reduce argmax across the 16 lanes of each half (butterfly) ----
#pragma unroll
    for (int r = 0; r < 8; ++r) {
#pragma unroll
        for (int mask = 1; mask < 16; mask <<= 1) {
            float ov = __shfl_xor(best[r], mask, 32);
            int   oi = __shfl_xor(bidx[r], mask, 32);
            bool take = (ov > best[r]) | ((ov == best[r]) & (oi < bidx[r]));
            best[r] = take ? ov : best[r];
            bidx[r] = take ? oi : bidx[r];
        }
    }

    // ---- write idx and gather v rows (row m handled by its own 16-lane half) ----
    const float* vh = v + (size_t)h * C_ * d_;
#pragma unroll
    for (int r = 0; r < 8; ++r) {
        int m  = r + half * 8;
        int ri = bidx[r];                       // uniform within the half
        if (l15 == 0)
            idx_out[(size_t)(b * H_ + h) * N_ + n0 + m] = ri;
        float4 vv = *(const float4*)(vh + (size_t)ri * d_ + l15 * 4);
        *(float4*)(out + ((size_t)(b * N_ + n0 + m)) * D_ + h * d_ + l15 * 4) = vv;
    }
}

extern "C" void kernel_launch(void* const* d_in, const int* in_sizes, int n_in,
                              void* d_out, int out_size, void* d_ws, size_t ws_size,
                              hipStream_t stream) {
    const float* x      = (const float*)d_in[0];
    const float* codes  = (const float*)d_in[1];
    const float* wk     = (const float*)d_in[2];
    const float* wv     = (const float*)d_in[3];
    const float* gumbel = (const float*)d_in[4];

    float* out     = (float*)d_out;
    int*   idx_out = (int*)((float*)d_out + (size_t)B_ * N_ * D_);

    float* kp = (float*)d_ws;                         // 4 MB
    float* v  = kp + (size_t)H_ * d_ * C_;            // 4 MB

    memcodes_kv_kernel<<<(H_ * C_ * d_) / 256, 256, 0, stream>>>(codes, wk, wv, kp, v);

    const int tiles = B_ * H_ * (N_ / 16);            // 4096
    memcodes_main_kernel<<<tiles / WAVES_PER_BLOCK, WAVES_PER_BLOCK * 32, 0, stream>>>(
        x, gumbel, kp, v, out, idx_out);
}